// STGSN_63737314672988
// MI455X (gfx1250) — compile-verified
//
#include <hip/hip_runtime.h>
#include <math.h>

#define NNODES 8192
#define TT 4
#define EE 262144
#define FF 128
#define HH 256

typedef __attribute__((ext_vector_type(2))) float v2f;
typedef __attribute__((ext_vector_type(8))) float v8f;

// ---------------------------------------------------------------------------
// GEMM via V_WMMA_F32_16X16X4_F32 (full fp32 matrix path on CDNA5).
// Y[M x Nn] = alpha * (X[M x K] @ B) + bias,  B = W (K x Nn) or W^T (Nn x K).
// One wave computes a 16x64 strip of D: 4 accumulators share one A fragment,
// so each A-frag load feeds 4 v_wmma issues (A is the streaming operand;
// W is L2-resident). TRANSB is a template arg -> branch-free inner loop.
// A frag: lane L holds A[tm*16 + L%16][k0 + (L/16)*2 .. +1]
// B frag: lane L holds B[k0 + (L/16)*2 .. +1][tile*16 + L%16]
// D frag: VGPR r, lane L -> D[tm*16 + r + (L/16)*8][tile*16 + L%16]
// ---------------------------------------------------------------------------
template <int TRANSB>
__global__ void gemm_wmma_f32(const float* __restrict__ X, const float* __restrict__ W,
                              const float* __restrict__ bias, float* __restrict__ Y,
                              int M, int K, int Nn, float alpha)
{
    const int lane = threadIdx.x & 31;
    const int wave = (int)((blockIdx.x * blockDim.x + threadIdx.x) >> 5);
    const int stripsN = Nn >> 6;                   // 64-wide strips (Nn=256 -> 4)
    const int totalStrips = (M >> 4) * stripsN;
    if (wave >= totalStrips) return;               // whole-wave uniform: EXEC stays all-1s
    const int tm = wave / stripsN;
    const int ts = wave - tm * stripsN;
    const int mn = lane & 15;
    const int m  = (tm << 4) + mn;                 // A row for this lane
    const int n0 = (ts << 6) + mn;                 // lane's col in tile j is n0 + 16*j
    const int khalf = (lane >> 4) << 1;            // 0 or 2

    const float* Xrow = X + (size_t)m * K;
    // Warm each lane's X row into cache (gfx1250 global_prefetch_b8 path).
    for (int p = 0; p < K; p += 32) __builtin_prefetch(Xrow + p, 0, 0);

    v8f acc0 = {}, acc1 = {}, acc2 = {}, acc3 = {};
    for (int k0 = 0; k0 < K; k0 += 4) {
        const int ka = k0 + khalf;
        v2f a;
        a.x = Xrow[ka];
        a.y = Xrow[ka + 1];
        v2f b0, b1, b2, b3;
        if (TRANSB) {
            const float* c0 = W + (size_t)(n0 +  0) * K + ka;
            const float* c1 = W + (size_t)(n0 + 16) * K + ka;
            const float* c2 = W + (size_t)(n0 + 32) * K + ka;
            const float* c3 = W + (size_t)(n0 + 48) * K + ka;
            b0.x = c0[0]; b0.y = c0[1];
            b1.x = c1[0]; b1.y = c1[1];
            b2.x = c2[0]; b2.y = c2[1];
            b3.x = c3[0]; b3.y = c3[1];
        } else {
            const float* r0 = W + (size_t)ka * Nn + n0;        // row ka
            const float* r1 = r0 + Nn;                         // row ka+1
            b0.x = r0[0];  b0.y = r1[0];
            b1.x = r0[16]; b1.y = r1[16];
            b2.x = r0[32]; b2.y = r1[32];
            b3.x = r0[48]; b3.y = r1[48];
        }
        acc0 = __builtin_amdgcn_wmma_f32_16x16x4_f32(false, a, false, b0, (short)0, acc0, false, false);
        acc1 = __builtin_amdgcn_wmma_f32_16x16x4_f32(false, a, false, b1, (short)0, acc1, false, false);
        acc2 = __builtin_amdgcn_wmma_f32_16x16x4_f32(false, a, false, b2, (short)0, acc2, false, false);
        acc3 = __builtin_amdgcn_wmma_f32_16x16x4_f32(false, a, false, b3, (short)0, acc3, false, false);
    }

    const int rbase = (tm << 4) + ((lane >> 4) << 3);
    v8f accs[4] = {acc0, acc1, acc2, acc3};
    #pragma unroll
    for (int j = 0; j < 4; ++j) {
        const int col = n0 + (j << 4);
        const float bv = bias ? bias[col] : 0.0f;
        #pragma unroll
        for (int r = 0; r < 8; ++r)
            Y[(size_t)(rbase + r) * Nn + col] = alpha * accs[j][r] + bv;
    }
}

// ---------------------------------------------------------------------------
// Degree construction: deg = 1 (self-loop) + sum of edge weights at src.
// ---------------------------------------------------------------------------
__global__ void fill_f32(float* __restrict__ p, long n, float v)
{
    long g = (long)blockIdx.x * blockDim.x + threadIdx.x;
    if (g < n) p[g] = v;
}

__global__ void deg_accum(const int* __restrict__ src,
                          float* __restrict__ degS, float* __restrict__ degG,
                          float4 coef)
{
    long g = (long)blockIdx.x * blockDim.x + threadIdx.x;
    if (g >= (long)TT * EE) return;
    int t = (int)(g / EE);
    int s = src[g];
    float c = (t == 0) ? coef.x : (t == 1) ? coef.y : (t == 2) ? coef.z : coef.w;
    atomicAdd(&degS[t * NNODES + s], 1.0f);
    atomicAdd(&degG[s], c);
}

// OUT[r][:] = XW[r][:] / deg[r]    (self-loop contribution: val = 1/deg)
__global__ void scale_rows(const float* __restrict__ XW, const float* __restrict__ deg,
                           float* __restrict__ OUT, long rows)
{
    long g = (long)blockIdx.x * blockDim.x + threadIdx.x;
    long total = rows * (HH / 4);
    if (g >= total) return;
    long r = g >> 6;
    int  h = ((int)g & 63) << 2;
    float inv = 1.0f / deg[r];
    float4 x = *(const float4*)&XW[r * HH + h];
    float4 o; o.x = inv * x.x; o.y = inv * x.y; o.z = inv * x.z; o.w = inv * x.w;
    *(float4*)&OUT[r * HH + h] = o;
}

// Edge scatter: OUT[row(src)] += (coef[t]/deg[src]) * XW[row(dst)]
// rowStride/degStride = NNODES for the per-t graphs, 0 for the global graph.
__global__ void spmm_scatter(const int* __restrict__ src, const int* __restrict__ dst,
                             const float* __restrict__ deg, const float* __restrict__ XW,
                             float* __restrict__ OUT, int rowStride, int degStride,
                             float4 coef)
{
    long g = (long)blockIdx.x * blockDim.x + threadIdx.x;
    const long total = (long)TT * EE * (HH / 4);
    if (g >= total) return;
    int e = (int)(g >> 6);
    int h = ((int)g & 63) << 2;
    int t = e / EE;
    int s = src[e];
    int d = dst[e];
    float c = (t == 0) ? coef.x : (t == 1) ? coef.y : (t == 2) ? coef.z : coef.w;
    float val = c / deg[(size_t)t * degStride + s];
    const float4 x = *(const float4*)&XW[((size_t)t * rowStride + d) * HH + h];
    float* o = OUT + ((size_t)t * rowStride + s) * HH + h;
    atomicAdd(o + 0, val * x.x);
    atomicAdd(o + 1, val * x.y);
    atomicAdd(o + 2, val * x.z);
    atomicAdd(o + 3, val * x.w);
}

__global__ void bias_relu(float* __restrict__ Y, const float* __restrict__ b, long rows)
{
    long g = (long)blockIdx.x * blockDim.x + threadIdx.x;
    long total = rows * (HH / 4);
    if (g >= total) return;
    long r = g >> 6;
    int  h = ((int)g & 63) << 2;
    float4 v = *(float4*)&Y[r * HH + h];
    v.x = fmaxf(v.x + b[h + 0], 0.0f);
    v.y = fmaxf(v.y + b[h + 1], 0.0f);
    v.z = fmaxf(v.z + b[h + 2], 0.0f);
    v.w = fmaxf(v.w + b[h + 3], 0.0f);
    *(float4*)&Y[r * HH + h] = v;
}

// ---------------------------------------------------------------------------
// Attention over T + decoder scores. One wave per node.
// e[t] = (emb[t,n] . qk[n]) / sqrt(H); a = softmax_t(e);
// agg[h] = sum_t a[t]*emb[t,n,h]; dyn = [agg, gbl];
// s1[n] = dyn . dec_w[:2H], s2[n] = dyn . dec_w[2H:]
// ---------------------------------------------------------------------------
__global__ void attn_scores(const float* __restrict__ emb, const float* __restrict__ qk,
                            const float* __restrict__ gbl, const float* __restrict__ decw,
                            float* __restrict__ s1, float* __restrict__ s2)
{
    const int lane = threadIdx.x & 31;
    const int node = (int)((blockIdx.x * blockDim.x + threadIdx.x) >> 5);
    if (node >= NNODES) return;

    float q[8], em[TT][8];
    #pragma unroll
    for (int i = 0; i < 8; ++i) {
        int h = lane + i * 32;
        q[i] = qk[(size_t)node * HH + h];
        #pragma unroll
        for (int t = 0; t < TT; ++t)
            em[t][i] = emb[((size_t)t * NNODES + node) * HH + h];
    }
    float e[TT];
    #pragma unroll
    for (int t = 0; t < TT; ++t) {
        float v = 0.0f;
        #pragma unroll
        for (int i = 0; i < 8; ++i) v += em[t][i] * q[i];
        for (int off = 16; off >= 1; off >>= 1) v += __shfl_xor(v, off, 32);
        e[t] = v * (1.0f / 16.0f);              // 1/sqrt(256)
    }
    float mx = fmaxf(fmaxf(e[0], e[1]), fmaxf(e[2], e[3]));
    float a[TT], ssum = 0.0f;
    #pragma unroll
    for (int t = 0; t < TT; ++t) { a[t] = expf(e[t] - mx); ssum += a[t]; }
    float inv = 1.0f / ssum;

    float p1 = 0.0f, p2 = 0.0f;
    #pragma unroll
    for (int i = 0; i < 8; ++i) {
        int h = lane + i * 32;
        float agg = 0.0f;
        #pragma unroll
        for (int t = 0; t < TT; ++t) agg += a[t] * inv * em[t][i];
        float gv = gbl[(size_t)node * HH + h];
        p1 += agg * decw[h]          + gv * decw[HH + h];
        p2 += agg * decw[2 * HH + h] + gv * decw[3 * HH + h];
    }
    for (int off = 16; off >= 1; off >>= 1) {
        p1 += __shfl_xor(p1, off, 32);
        p2 += __shfl_xor(p2, off, 32);
    }
    if (lane == 0) { s1[node] = p1; s2[node] = p2; }
}

// adj[i,j] = sigmoid(s2[i] + s1[j] + dec_b) -- 256MB streaming write.
__global__ void adj_sigmoid(const float* __restrict__ s1, const float* __restrict__ s2,
                            const float* __restrict__ decb, float* __restrict__ out)
{
    long g = (long)blockIdx.x * blockDim.x + threadIdx.x;
    const long total = (long)NNODES * (NNODES / 4);
    if (g >= total) return;
    int i = (int)(g / (NNODES / 4));
    int j = ((int)(g % (NNODES / 4))) * 4;
    float base = s2[i] + decb[0];
    const float a = s1[j], b = s1[j + 1], c = s1[j + 2], d = s1[j + 3];
    float4 r;
    r.x = 1.0f / (1.0f + expf(-(base + a)));
    r.y = 1.0f / (1.0f + expf(-(base + b)));
    r.z = 1.0f / (1.0f + expf(-(base + c)));
    r.w = 1.0f / (1.0f + expf(-(base + d)));
    *(float4*)&out[(size_t)i * NNODES + j] = r;
}

// ---------------------------------------------------------------------------
extern "C" void kernel_launch(void* const* d_in, const int* in_sizes, int n_in,
                              void* d_out, int out_size, void* d_ws, size_t ws_size,
                              hipStream_t stream)
{
    (void)in_sizes; (void)n_in; (void)out_size; (void)ws_size;
    const int*   e_src = (const int*)  d_in[0];
    const int*   e_dst = (const int*)  d_in[1];
    const float* feats = (const float*)d_in[2];
    const float* W0    = (const float*)d_in[3];
    const float* b0    = (const float*)d_in[4];
    const float* W1    = (const float*)d_in[5];
    const float* b1    = (const float*)d_in[6];
    const float* mlpW  = (const float*)d_in[7];
    const float* mlpb  = (const float*)d_in[8];
    const float* Wq    = (const float*)d_in[9];
    const float* Wk    = (const float*)d_in[10];
    const float* decw  = (const float*)d_in[11];
    const float* decb  = (const float*)d_in[12];
    float* out = (float*)d_out;

    // workspace carve-up (floats)
    float* ws = (float*)d_ws;
    size_t off = 0;
    const size_t TNH = (size_t)TT * NNODES * HH;
    const size_t NH  = (size_t)NNODES * HH;
    float* BUF_A = ws + off; off += TNH;     // x@W scratch / ind_emb
    float* BUF_B = ws + off; off += TNH;     // ind features
    float* G_TMP = ws + off; off += NH;      // gbl x@W scratch
    float* G_OUT = ws + off; off += NH;      // gbl
    float* QBUF  = ws + off; off += NH;      // q
    float* QK    = ws + off; off += NH;      // q @ Wk^T
    float* DEG_S = ws + off; off += (size_t)TT * NNODES;
    float* DEG_G = ws + off; off += NNODES;
    float* S1    = ws + off; off += NNODES;
    float* S2    = ws + off; off += NNODES;

    // temporal coefficients: (1-THETA)^(tau - t) normalized
    float c3 = 0.9f, c2 = c3 * 0.9f, c1 = c2 * 0.9f, c0 = c1 * 0.9f;
    float cs = c0 + c1 + c2 + c3;
    float4 coefG = make_float4(c0 / cs, c1 / cs, c2 / cs, c3 / cs);
    float4 coef1 = make_float4(1.0f, 1.0f, 1.0f, 1.0f);

    auto gemm = [&](const float* X, const float* W, const float* bias, float* Y,
                    int M, int K, int Nn, float alpha) {
        int strips = (M / 16) * (Nn / 64);
        int thr = strips * 32;
        gemm_wmma_f32<0><<<(thr + 127) / 128, 128, 0, stream>>>(X, W, bias, Y, M, K, Nn, alpha);
    };
    auto gemmT = [&](const float* X, const float* W, float* Y, int M, int K, int Nn) {
        int strips = (M / 16) * (Nn / 64);
        int thr = strips * 32;
        gemm_wmma_f32<1><<<(thr + 127) / 128, 128, 0, stream>>>(X, W, nullptr, Y, M, K, Nn, 1.0f);
    };
    auto rowgrid = [](long rows) { return dim3((unsigned)((rows * (HH / 4) + 255) / 256)); };

    const long TN = (long)TT * NNODES;
    const long edgeThr = (long)TT * EE * (HH / 4);

    // --- degrees (self-loop inclusive) ---
    fill_f32<<<(TN + 255) / 256, 256, 0, stream>>>(DEG_S, TN, 1.0f);
    fill_f32<<<(NNODES + 255) / 256, 256, 0, stream>>>(DEG_G, NNODES, 1.0f);
    deg_accum<<<((long)TT * EE + 255) / 256, 256, 0, stream>>>(e_src, DEG_S, DEG_G, coefG);

    // --- ind branch, layer 0 (cites == ind, folded out) ---
    gemm(feats, W0, nullptr, BUF_A, TT * NNODES, FF, HH, 1.0f);
    scale_rows<<<rowgrid(TN), 256, 0, stream>>>(BUF_A, DEG_S, BUF_B, TN);
    spmm_scatter<<<(edgeThr + 255) / 256, 256, 0, stream>>>(e_src, e_dst, DEG_S, BUF_A, BUF_B,
                                                            NNODES, NNODES, coef1);
    bias_relu<<<rowgrid(TN), 256, 0, stream>>>(BUF_B, b0, TN);
    // --- ind branch, layer 1 ---
    gemm(BUF_B, W1, nullptr, BUF_A, TT * NNODES, HH, HH, 1.0f);
    scale_rows<<<rowgrid(TN), 256, 0, stream>>>(BUF_A, DEG_S, BUF_B, TN);
    spmm_scatter<<<(edgeThr + 255) / 256, 256, 0, stream>>>(e_src, e_dst, DEG_S, BUF_A, BUF_B,
                                                            NNODES, NNODES, coef1);
    bias_relu<<<rowgrid(TN), 256, 0, stream>>>(BUF_B, b1, TN);

    // --- gbl branch on feats[-1] with global temporal graph ---
    const float* feats3 = feats + (size_t)3 * NNODES * FF;
    gemm(feats3, W0, nullptr, G_TMP, NNODES, FF, HH, 1.0f);
    scale_rows<<<rowgrid(NNODES), 256, 0, stream>>>(G_TMP, DEG_G, G_OUT, NNODES);
    spmm_scatter<<<(edgeThr + 255) / 256, 256, 0, stream>>>(e_src, e_dst, DEG_G, G_TMP, G_OUT,
                                                            0, 0, coefG);
    bias_relu<<<rowgrid(NNODES), 256, 0, stream>>>(G_OUT, b0, NNODES);
    gemm(G_OUT, W1, nullptr, G_TMP, NNODES, HH, HH, 1.0f);
    scale_rows<<<rowgrid(NNODES), 256, 0, stream>>>(G_TMP, DEG_G, G_OUT, NNODES);
    spmm_scatter<<<(edgeThr + 255) / 256, 256, 0, stream>>>(e_src, e_dst, DEG_G, G_TMP, G_OUT,
                                                            0, 0, coefG);
    bias_relu<<<rowgrid(NNODES), 256, 0, stream>>>(G_OUT, b1, NNODES);  // G_OUT = gbl

    // --- ind_emb = (ind + cites) @ mlpW + mlpb = (2*ind) @ mlpW + mlpb ---
    gemm(BUF_B, mlpW, mlpb, BUF_A, TT * NNODES, HH, HH, 2.0f);

    // --- q = gbl @ Wq ; qk = q @ Wk^T (so k never materializes) ---
    gemm(G_OUT, Wq, nullptr, QBUF, NNODES, HH, HH, 1.0f);
    gemmT(QBUF, Wk, QK, NNODES, HH, HH);

    // --- softmax attention + decoder score vectors ---
    attn_scores<<<NNODES / 8, 256, 0, stream>>>(BUF_A, QK, G_OUT, decw, S1, S2);

    // --- adj[i,j] = sigmoid(s2[i] + s1[j] + dec_b) ---
    adj_sigmoid<<<((long)NNODES * (NNODES / 4) + 255) / 256, 256, 0, stream>>>(S1, S2, decb, out);
}